// OriginalGraphConvolution_22368189677639
// MI455X (gfx1250) — compile-verified
//
#include <hip/hip_runtime.h>
#include <stdint.h>

typedef __attribute__((ext_vector_type(2))) float v2f;
typedef __attribute__((ext_vector_type(8))) float v8f;

#define GCN_N_NODES 100000
#define GCN_N_EDGES 1600000
#define GCN_D 128

// ---------------------------------------------------------------------------
// Kernel 0: out[n, f] = bias[f]   (also zero-inits the poisoned output so the
// atomic scatter can accumulate on top).  float4 per thread.
// total float4 elements = 100000*128/4 = 3,200,000 -> 12,500 blocks of 256.
// ---------------------------------------------------------------------------
__global__ void __launch_bounds__(256)
gcn_bias_init_kernel(const float* __restrict__ bias, float* __restrict__ out) {
    const int t = blockIdx.x * 256 + threadIdx.x;
    const float4 b = ((const float4*)bias)[t & 31];   // row = 32 float4s wide
    ((float4*)out)[t] = b;
}

// ---------------------------------------------------------------------------
// Kernel 1: support = X @ W using V_WMMA_F32_16X16X4_F32 (fp32 matrix pipe).
//   block = 256 threads = 8 waves; block owns rows [m0, m0+16).
//   X tile staged to LDS via GLOBAL_LOAD_ASYNC_TO_LDS_B128 (ASYNCcnt path),
//   row pitch 132 floats (528 B, 16B-aligned) to kill bank conflicts.
//   wave w computes the 16x16 output tile at columns [16w, 16w+16).
// A-matrix VGPR layout (16x4 f32): lane m = lane&15, K = vgpr + 2*(lane>>4).
// C/D layout (16x16 f32):          N = lane&15, M = vgpr + 8*(lane>>4).
// ---------------------------------------------------------------------------
__global__ void __launch_bounds__(256)
gcn_gemm_wmma_f32_kernel(const float* __restrict__ X,
                         const float* __restrict__ W,
                         float* __restrict__ S) {
    __shared__ float ldsX[16 * 132];

    const int tid = threadIdx.x;
    const int m0  = blockIdx.x * 16;

    // ---- Async stage of the 16x128 X tile into padded LDS ----------------
    // 512 16-byte chunks (16 rows x 32 chunks); 256 threads x 2 iterations.
    // LDS address comes from the VDST VGPR: low 32 bits of the generic shared
    // pointer are the in-aperture LDS byte offset.
#pragma unroll
    for (int it = 0; it < 2; ++it) {
        const int chunk = tid + it * 256;            // 0..511
        const int r = chunk >> 5;                    // row    0..15
        const int c = chunk & 31;                    // 16B chunk 0..31
        const float*   gsrc    = X + (m0 + r) * GCN_D + c * 4;
        const unsigned lds_off = (unsigned)(unsigned long long)(&ldsX[r * 132 + c * 4]);
        asm volatile("global_load_async_to_lds_b128 %0, %1, off"
                     :: "v"(lds_off), "v"(gsrc)
                     : "memory");
    }
    asm volatile("s_wait_asynccnt 0x0" ::: "memory");
    __syncthreads();

    const int lane  = tid & 31;
    const int wave  = tid >> 5;       // 0..7 -> column tile
    const int half  = lane >> 4;      // 0 or 1 (selects K/M offset)
    const int l16   = lane & 15;
    const int n     = wave * 16 + l16;

    v8f acc = {};
#pragma unroll
    for (int k = 0; k < GCN_D; k += 4) {
        const int kk = k + 2 * half;
        // A: two consecutive K values for row l16 (8B-aligned LDS read).
        v2f a = *(const v2f*)&ldsX[l16 * 132 + kk];
        // B: W[kk][n], W[kk+1][n] — W is 64 KB, WGP$/L2 resident.
        v2f b;
        b.x = W[kk * GCN_D + n];
        b.y = W[(kk + 1) * GCN_D + n];
        acc = __builtin_amdgcn_wmma_f32_16x16x4_f32(
            /*neg_a=*/false, a, /*neg_b=*/false, b,
            /*c_mod=*/(short)0, acc, /*reuse_a=*/false, /*reuse_b=*/false);
    }

#pragma unroll
    for (int i = 0; i < 8; ++i) {
        S[(m0 + i + 8 * half) * GCN_D + n] = acc[i];
    }
}

// ---------------------------------------------------------------------------
// Kernel 2: COO scatter — one wave per edge, 4 floats per lane.
//   out[row, :] += val * support[col, :]
// float4 gather (L2-resident support) + 4 hardware fp32 atomics per lane.
// blocks of 256 threads handle 8 edges; 1.6M / 8 = 200,000 blocks exact.
// ---------------------------------------------------------------------------
__global__ void __launch_bounds__(256)
gcn_spmm_scatter_kernel(const int64_t* __restrict__ idx,
                        const float* __restrict__ vals,
                        const float* __restrict__ S,
                        float* __restrict__ out) {
    const int lane = threadIdx.x & 31;
    const int wave = threadIdx.x >> 5;
    const long long e = (long long)blockIdx.x * 8 + wave;

    const long long row = idx[e];
    const long long col = idx[(long long)GCN_N_EDGES + e];
    const float v = vals[e];

    const float4 s = *(const float4*)(S + col * GCN_D + lane * 4);
    float* dst = out + row * GCN_D + lane * 4;

    unsafeAtomicAdd(dst + 0, v * s.x);
    unsafeAtomicAdd(dst + 1, v * s.y);
    unsafeAtomicAdd(dst + 2, v * s.z);
    unsafeAtomicAdd(dst + 3, v * s.w);
}

// ---------------------------------------------------------------------------
// Host-side launcher.
// Inputs (setup_inputs order):
//   0: node_features float32 [100000,128]
//   1: weight        float32 [128,128]
//   2: bias          float32 [128]
//   3: adj_indices   int64   [2,1600000]
//   4: adj_values    float32 [1600000]
// d_out: float32 [100000,128].  d_ws: support scratch (51.2 MB needed).
// ---------------------------------------------------------------------------
extern "C" void kernel_launch(void* const* d_in, const int* in_sizes, int n_in,
                              void* d_out, int out_size, void* d_ws, size_t ws_size,
                              hipStream_t stream) {
    const float*   X    = (const float*)d_in[0];
    const float*   W    = (const float*)d_in[1];
    const float*   bias = (const float*)d_in[2];
    const int64_t* idx  = (const int64_t*)d_in[3];
    const float*   vals = (const float*)d_in[4];

    float* out     = (float*)d_out;
    float* support = (float*)d_ws;   // [100000, 128] fp32 = 51.2 MB

    // out = bias (broadcast): 3.2M float4s / 256 = 12,500 blocks.
    gcn_bias_init_kernel<<<(GCN_N_NODES * GCN_D / 4) / 256, 256, 0, stream>>>(bias, out);

    // support = X @ W : one block per 16 rows -> 6250 blocks of 8 waves.
    gcn_gemm_wmma_f32_kernel<<<GCN_N_NODES / 16, 256, 0, stream>>>(X, W, support);

    // scatter: 8 edges per block -> 200,000 blocks.
    gcn_spmm_scatter_kernel<<<GCN_N_EDGES / 8, 256, 0, stream>>>(idx, vals, support, out);
}